// EggSurfaceRenderer_67250597921248
// MI455X (gfx1250) — compile-verified
//
#include <hip/hip_runtime.h>
#include <hip/hip_bf16.h>

// ---------------------------------------------------------------------------
// EggSurfaceRenderer for gfx1250 (MI455X)
//   grid  = 2048 blocks (one per output row)
//   block = 256 threads (8 wave32s), each thread shades 8 pixels
//   Per row: v (texture row coordinate) is uniform -> stage the two needed
//   rows of each texture (4 x 12KB = 48KB) into LDS with the Tensor Data
//   Mover, then all bilinear taps are LDS gathers.
// ---------------------------------------------------------------------------

#define RES   2048
#define TW    1024           // texture width/height
#define ROWF  (3 * TW)       // floats per texture row = 3072
#define BLK   256

typedef __attribute__((ext_vector_type(4))) unsigned int v4u_t;
typedef __attribute__((ext_vector_type(8))) int          v8i_t;
typedef __attribute__((ext_vector_type(4))) int          v4i_t;

#if __has_builtin(__builtin_amdgcn_tensor_load_to_lds)
#define EGG_USE_TDM 1
#else
#define EGG_USE_TDM 0
#endif

__device__ __forceinline__ float clampf(float x, float lo, float hi) {
    return fminf(fmaxf(x, lo), hi);
}

#if EGG_USE_TDM
// Issue one TDM descriptor: copy one texture row (3072 dwords = 12KB) from
// global memory to LDS byte offset lds_off.  D# fields per CDNA5 ISA §8.
__device__ __forceinline__ void tdm_load_row(const float* gsrc, unsigned lds_off) {
    unsigned long long ga = (unsigned long long)(const void*)gsrc;
    v4u_t g0;
    g0.x = 1u;                                             // count=1, user desc
    g0.y = lds_off;                                        // lds_addr (bytes)
    g0.z = (unsigned)(ga & 0xFFFFFFFFu);                   // global_addr[31:0]
    g0.w = (unsigned)((ga >> 32) & 0x1FFFFFFu)             // global_addr[56:32]
         | (2u << 30);                                     // type = 2 (image)
    v8i_t g1;
    g1[0] = 0x00020000;      // workgroup_mask=0, data_size=2 (4B)
    g1[1] = 0x0C000000;      // tensor_dim0[15:0]=3072 in bits 63:48
    g1[2] = 0x00010000;      // tensor_dim0[31:16]=0, tensor_dim1[15:0]=1
    g1[3] = 0x0C000000;      // tensor_dim1[31:16]=0, tile_dim0=3072
    g1[4] = 0x00000001;      // tile_dim1=1, tile_dim2=0
    g1[5] = 3072;            // tensor_dim0_stride[31:0]
    g1[6] = 0;               // tensor_dim0_stride[47:32], dim1_stride lo
    g1[7] = 0;               // tensor_dim1_stride hi
    v4i_t z4 = {0, 0, 0, 0};
#if __has_include(<hip/amd_detail/amd_gfx1250_TDM.h>)
    v8i_t z8 = {0, 0, 0, 0, 0, 0, 0, 0};
    __builtin_amdgcn_tensor_load_to_lds(g0, g1, z4, z4, z8, 0);   // 6-arg form
#else
    __builtin_amdgcn_tensor_load_to_lds(g0, g1, z4, z4, 0);       // 5-arg form
#endif
}
#endif

__global__ __launch_bounds__(BLK)
void egg_render_kernel(const float* __restrict__ tex,
                       const float* __restrict__ nmap,
                       float* __restrict__ out) {
    __shared__ float lds[4 * ROWF];          // 48 KB: texY0, texY1, nmY0, nmY1

    const int py  = blockIdx.x;              // output row
    const int tid = threadIdx.x;

    const float inv_scale2 = 2.0f / 0.85f;
    const float ny = (0.5f - (float)py / (float)RES) * inv_scale2;

    // ---- per-row uniform texture row selection (v depends only on ny) ----
    const float phi = acosf(clampf(ny, -1.0f, 1.0f));
    float v = phi * 0.31830988618530718f;    // phi / pi
    v = v - floorf(v);                       // v % 1.0 (sampler wrap)
    const float yy = v * (float)TW - 0.5f;
    const float yf = floorf(yy);
    const int   y0 = ((int)yf) & (TW - 1);
    const int   y1 = (y0 + 1) & (TW - 1);
    const float fy = yy - yf;

    // ---- stage 4 texture rows into LDS ----
#if EGG_USE_TDM
    if (tid < 32) {                          // one wave issues the DMA
        tdm_load_row(tex  + (size_t)y0 * ROWF, (unsigned)(size_t)(lds + 0 * ROWF));
        tdm_load_row(tex  + (size_t)y1 * ROWF, (unsigned)(size_t)(lds + 1 * ROWF));
        tdm_load_row(nmap + (size_t)y0 * ROWF, (unsigned)(size_t)(lds + 2 * ROWF));
        tdm_load_row(nmap + (size_t)y1 * ROWF, (unsigned)(size_t)(lds + 3 * ROWF));
        __builtin_amdgcn_s_wait_tensorcnt(0);
    }
#else
    {
        const float* rows[4] = { tex  + (size_t)y0 * ROWF, tex  + (size_t)y1 * ROWF,
                                 nmap + (size_t)y0 * ROWF, nmap + (size_t)y1 * ROWF };
        for (int r = 0; r < 4; ++r) {
            const float4* s = reinterpret_cast<const float4*>(rows[r]);
            float4*       d = reinterpret_cast<float4*>(lds + r * ROWF);
            for (int k = tid; k < ROWF / 4; k += BLK) d[k] = s[k];
        }
    }
#endif
    __syncthreads();

    const float* tr0 = lds + 0 * ROWF;
    const float* tr1 = lds + 1 * ROWF;
    const float* nr0 = lds + 2 * ROWF;
    const float* nr1 = lds + 3 * ROWF;

    // ---- lighting constants (compile-time folded) ----
    const float ll = sqrtf(0.5f * 0.5f + 0.7f * 0.7f + 1.0f * 1.0f);
    const float LX = 0.5f / ll, LY = 0.7f / ll, LZ = 1.0f / ll;
    const float hl = sqrtf(LX * LX + LY * LY + (LZ + 1.0f) * (LZ + 1.0f));
    const float HX = LX / hl, HY = LY / hl, HZ = (LZ + 1.0f) / hl;
    const float F0 = 0.04f, BUMP = 1.0f, AMB = 0.1f;

    const float egg  = 1.0f - 0.25f * ny;
    const bool  safe = fabsf(egg) > 0.01f;

    for (int px = tid; px < RES; px += BLK) {
        const float nx = ((float)px / (float)RES - 0.5f) * inv_scale2;
        const float r2 = nx * nx + ny * ny;
        const float z  = sqrtf(fmaxf(1.0f - r2, 0.0f));

        const float x_egg = safe ? nx / egg : nx;
        const float z_egg = safe ? z / egg  : z;

        const float theta = atan2f(z_egg, x_egg);
        float u = theta * 0.15915494309189535f + 0.5f;   // theta/(2pi) + 0.5
        u = u - floorf(u);

        // ---- bilinear taps from LDS-staged rows ----
        const float x  = u * (float)TW - 0.5f;
        const float xf = floorf(x);
        const int   x0 = ((int)xf) & (TW - 1);
        const int   x1 = (x0 + 1) & (TW - 1);
        const float fx = x - xf;
        const float w00 = (1.0f - fx) * (1.0f - fy);
        const float w01 = fx * (1.0f - fy);
        const float w10 = (1.0f - fx) * fy;
        const float w11 = fx * fy;

        float alb[3], tn[3];
#pragma unroll
        for (int c = 0; c < 3; ++c) {
            alb[c] = tr0[x0 * 3 + c] * w00 + tr0[x1 * 3 + c] * w01
                   + tr1[x0 * 3 + c] * w10 + tr1[x1 * 3 + c] * w11;
            const float s = nr0[x0 * 3 + c] * w00 + nr0[x1 * 3 + c] * w01
                          + nr1[x0 * 3 + c] * w10 + nr1[x1 * 3 + c] * w11;
            tn[c] = s * 2.0f - 1.0f;
        }

        // ---- analytic egg normal + TBN ----
        const float snx = nx * egg, sny = ny, snz = z * egg;
        const float sninv = 1.0f / fmaxf(sqrtf(snx * snx + sny * sny + snz * snz), 1e-6f);
        const float bx = snx * sninv, by = sny * sninv, bz = snz * sninv;

        const bool  up  = fabsf(by) > 0.99f;
        const float rx  = up ? 1.0f : 0.0f;
        const float ry  = up ? 0.0f : 1.0f;
        const float d   = rx * bx + ry * by;          // ref.z == 0
        float tx = rx - d * bx, ty = ry - d * by, tz = -d * bz;
        const float tinv = 1.0f / fmaxf(sqrtf(tx * tx + ty * ty + tz * tz), 1e-6f);
        tx *= tinv; ty *= tinv; tz *= tinv;
        const float cbx = by * tz - bz * ty;          // bitangent = n x t
        const float cby = bz * tx - bx * tz;
        const float cbz = bx * ty - by * tx;

        // ---- perturb + renormalize ----
        float wx = tn[0] * tx + tn[1] * cbx + tn[2] * bx;
        float wy = tn[0] * ty + tn[1] * cby + tn[2] * by;
        float wz = tn[0] * tz + tn[1] * cbz + tn[2] * bz;
        wx = bx * (1.0f - BUMP) + wx * BUMP;
        wy = by * (1.0f - BUMP) + wy * BUMP;
        wz = bz * (1.0f - BUMP) + wz * BUMP;
        const float ninv = 1.0f / fmaxf(sqrtf(wx * wx + wy * wy + wz * wz), 1e-6f);
        wx *= ninv; wy *= ninv; wz *= ninv;

        // ---- Blinn-Phong + Schlick ----
        const float ndl = fmaxf(wx * LX + wy * LY + wz * LZ, 0.0f);
        const float ndh = fmaxf(wx * HX + wy * HY + wz * HZ, 0.0f);
        const float s2 = ndh * ndh, s4 = s2 * s2, s8 = s4 * s4, s16 = s8 * s8;
        const float spec = s16 * s16;                 // ndh^32
        const float ndv = fmaxf(wz, 0.0f);
        const float om = 1.0f - ndv;
        const float om2 = om * om;
        const float fres = F0 + (1.0f - F0) * (om2 * om2 * om);
        const float diff = AMB + ndl;

        const bool inside = (r2 <= 1.0f);
        float* o = out + ((size_t)py * RES + px) * 3;
#pragma unroll
        for (int c = 0; c < 3; ++c) {
            const float col = clampf(alb[c] * diff + spec * fres, 0.0f, 1.0f);
            o[c] = inside ? col : 0.0f;
        }
    }
}

extern "C" void kernel_launch(void* const* d_in, const int* in_sizes, int n_in,
                              void* d_out, int out_size, void* d_ws, size_t ws_size,
                              hipStream_t stream) {
    (void)in_sizes; (void)n_in; (void)out_size; (void)d_ws; (void)ws_size;
    const float* tex  = (const float*)d_in[0];   // (1024,1024,3) f32
    const float* nmap = (const float*)d_in[1];   // (1024,1024,3) f32
    float*       out  = (float*)d_out;           // (2048,2048,3) f32
    egg_render_kernel<<<RES, BLK, 0, stream>>>(tex, nmap, out);
}